// ColorFeatureExtractor_14353780703860
// MI455X (gfx1250) — compile-verified
//
#include <hip/hip_runtime.h>
#include <math.h>

#define KC 5
#define NPIX (224 * 224)
#define MAXIT 100
#define WAVES 32                    // 1024 threads per workgroup
#define TPW (NPIX / 16 / WAVES)     // 98 16-pixel tiles per wave

static_assert(NPIX % (16 * WAVES) == 0, "tile partition must be exact");

typedef float v2f __attribute__((ext_vector_type(2)));
typedef float v8f __attribute__((ext_vector_type(8)));

// One workgroup per batch image. Entire K-means iteration loop stays on-chip:
// centers in LDS, per-wave partial sums in LDS, workgroup barriers between
// phases. Score matrix (16 pixels x 5 centers) per wave-tile comes from a
// single V_WMMA_F32_16X16X4_F32: D = A(16x4 pixels) * B(4x16 = -2*centers) + C(c^2).
__global__ __launch_bounds__(1024) void kmeans_wmma_kernel(
    const float* __restrict__ x, const int* __restrict__ init_idx,
    float* __restrict__ colorinfo)
{
  const int b = blockIdx.x;
  const float* __restrict__ px = x + (size_t)b * 3 * NPIX;   // planar RGB

  __shared__ float s_centers[KC][3];
  __shared__ float s_sums[WAVES][KC][4];   // per-wave {count, r, g, b}
  __shared__ float s_tot[KC][4];
  __shared__ float s_scores[WAVES][16][KC];
  __shared__ int   s_conv;

  const int tid  = threadIdx.x;
  const int wave = tid >> 5;
  const int lane = tid & 31;
  const int half = lane >> 4;   // 0: A holds K=0,1 (r,g)   1: K=2,3 (b,pad)
  const int nloc = lane & 15;   // A row (pixel) / B,C,D column (center) index
  const int base = wave * TPW * 16 + nloc;

  // centers0 = pixels[:, init_idx] (+1e-8 like the reference pixels)
  if (tid < KC * 3) {
    int k = tid / 3, d = tid % 3;
    int p = init_idx[k];
    s_centers[k][d] = px[(size_t)d * NPIX + p] + 1e-8f;
  }
  __syncthreads();

  for (int it = 0; it < MAXIT; ++it) {
    if (lane < KC * 4) s_sums[wave][lane >> 2][lane & 3] = 0.0f;

    // ---- B fragment: B[k][n] = -2 * center_n[k] (k=3 row and n>=5 cols = 0)
    // ---- C fragment: C[m][n] = |center_n|^2  (1e30 for padding columns)
    v2f bfrag; float c2v;
    if (nloc < KC) {
      float cr = s_centers[nloc][0];
      float cg = s_centers[nloc][1];
      float cb = s_centers[nloc][2];
      c2v = cr * cr + cg * cg + cb * cb;
      if (half == 0) { bfrag.x = -2.0f * cr; bfrag.y = -2.0f * cg; }
      else           { bfrag.x = -2.0f * cb; bfrag.y = 0.0f; }
    } else {
      bfrag.x = 0.0f; bfrag.y = 0.0f;
      c2v = 1e30f;                      // padding columns never win argmin
    }
    v8f cfrag;
#pragma unroll
    for (int v = 0; v < 8; ++v) cfrag[v] = c2v;

    // ---- assignment pass over this wave's 98 tiles of 16 pixels.
    // Loads are UNIFORM across the wave (every lane reads r,g,b; selects by
    // half with cndmask) so there is no exec divergence around vmem and the
    // scheduler can overlap loads of tile t+1 with LDS work of tile t.
#pragma unroll 2
    for (int t = 0; t < TPW; ++t) {
      const int p = base + t * 16;
      const float r  = px[p] + 1e-8f;
      const float g  = px[NPIX + p] + 1e-8f;
      const float bl = px[2 * NPIX + p] + 1e-8f;
      v2f afrag;                         // A[m][k]: m=nloc, k-pair per half
      afrag.x = half ? bl   : r;
      afrag.y = half ? 0.0f : g;

      // score[m][n] = c2_n - 2 * x_m . c_n   (argmin-equivalent to d^2)
      v8f d = __builtin_amdgcn_wmma_f32_16x16x4_f32(
          false, afrag, false, bfrag, (short)0, cfrag, false, false);

      // spill the 5 useful columns: VGPR v in lane L holds (m = v+8*half, n = nloc)
      if (nloc < KC) {
#pragma unroll
        for (int v = 0; v < 8; ++v)
          s_scores[wave][v + 8 * half][nloc] = d[v];
      }

      // lanes 0..15: argmin over the 5 centers for pixel nloc, accumulate
      if (half == 0) {
        float best = s_scores[wave][nloc][0];
        int bi = 0;
#pragma unroll
        for (int k = 1; k < KC; ++k) {
          float s = s_scores[wave][nloc][k];
          if (s < best) { best = s; bi = k; }
        }
        atomicAdd(&s_sums[wave][bi][0], 1.0f);
        atomicAdd(&s_sums[wave][bi][1], r);
        atomicAdd(&s_sums[wave][bi][2], g);
        atomicAdd(&s_sums[wave][bi][3], bl);
      }
    }
    __syncthreads();

    // ---- cross-wave reduction of per-cluster sums
    if (tid < KC * 4) {
      int k = tid >> 2, c = tid & 3;
      float acc = 0.0f;
      for (int w = 0; w < WAVES; ++w) acc += s_sums[w][k][c];
      s_tot[k][c] = acc;
    }
    __syncthreads();

    // ---- centers update + convergence (jax.lax.scan carry semantics:
    // on convergence keep OLD centers; percentages from this assignment)
    if (tid == 0) {
      int conv = 1;
      float newc[KC][3];
      for (int k = 0; k < KC; ++k) {
        float cnt = s_tot[k][0];
        for (int d = 0; d < 3; ++d) {
          float nc = s_tot[k][d + 1] / cnt;
          newc[k][d] = nc;
          float diff = fabsf(s_centers[k][d] - nc);
          float tol  = 1e-8f + 1e-5f * fabsf(nc);
          if (!(diff <= tol)) conv = 0;
        }
      }
      s_conv = conv;
      if (!conv)
        for (int k = 0; k < KC; ++k)
          for (int d = 0; d < 3; ++d) s_centers[k][d] = newc[k][d];
    }
    __syncthreads();
    if (s_conv) break;
  }

  // color_info[b] = concat(clip(centers), clip(counts/N)), then nan_to_num
  if (tid < KC * 4) {
    int k = tid >> 2, c = tid & 3;
    float v = (c < 3) ? s_centers[k][c] : (s_tot[k][0] * (1.0f / NPIX));
    v = (v != v) ? 0.0f : fminf(fmaxf(v, 0.0f), 1.0f); // clip maps +/-inf to 1/0
    colorinfo[(b * KC + k) * 4 + c] = v;
  }
}

// Reproduces the reference's color_info[sort_idx] fancy-indexing over the
// BATCH axis: out[b,k,:, :] = color_info[sort_idx[b,k], :, :], shape (8,5,5,4).
__global__ void sort_gather_kernel(const float* __restrict__ ci,
                                   float* __restrict__ out)
{
  __shared__ float s_ci[8 * KC * 4];
  __shared__ int   s_idx[8][KC];
  const int tid = threadIdx.x;

  for (int e = tid; e < 8 * KC * 4; e += blockDim.x) s_ci[e] = ci[e];
  __syncthreads();

  // stable descending argsort of percentages via rank counting
  if (tid < 8 * KC) {
    int bb = tid / KC, i = tid % KC;
    float pi = s_ci[(bb * KC + i) * 4 + 3];
    int r = 0;
    for (int j = 0; j < KC; ++j) {
      float pj = s_ci[(bb * KC + j) * 4 + 3];
      if (pj > pi || (pj == pi && j < i)) ++r;
    }
    s_idx[bb][r] = i;
  }
  __syncthreads();

  for (int e = tid; e < 8 * KC * KC * 4; e += blockDim.x) {
    int c  = e & 3;
    int k2 = (e >> 2) % KC;
    int k  = (e / (4 * KC)) % KC;
    int bb = e / (4 * KC * KC);
    out[e] = s_ci[(s_idx[bb][k] * KC + k2) * 4 + c];  // batch-axis gather
  }
}

extern "C" void kernel_launch(void* const* d_in, const int* in_sizes, int n_in,
                              void* d_out, int out_size, void* d_ws, size_t ws_size,
                              hipStream_t stream) {
  (void)in_sizes; (void)n_in; (void)out_size; (void)ws_size;
  const float* x        = (const float*)d_in[0];  // (8,3,224,224) f32
  const int*   init_idx = (const int*)d_in[1];    // (5,) i32
  float* ci  = (float*)d_ws;                      // color_info scratch (8*5*4)
  float* out = (float*)d_out;                     // (8,5,5,4) f32

  kmeans_wmma_kernel<<<8, 1024, 0, stream>>>(x, init_idx, ci);
  sort_gather_kernel<<<1, 256, 0, stream>>>(ci, out);
}